// DPM_Block_19241453486889
// MI455X (gfx1250) — compile-verified
//
#include <hip/hip_runtime.h>
#include <hip/hip_bf16.h>

// MI455X (gfx1250) implementation. B=128, S=49, C=512, L=1000.
// - fp32 WMMA (v_wmma_f32_16x16x4_f32) for all GEMMs (exact fp32 path).
// - Attention fully fused: no (B,L,C)=262MB intermediates; Fs_p[b] staged into
//   LDS via the Tensor Data Mover (tensor_load_to_lds + s_wait_tensorcnt).
// - Conv GEMM: activations and weight K-panels both staged in LDS (stride 516
//   dwords -> conflict-free strided WMMA-operand reads, 16B-aligned rows).
// Workspace need: ~34.5 MB of d_ws.

#define B_ 128
#define S_ 49
#define C_ 512
#define L_ 1000

typedef __attribute__((ext_vector_type(2))) float v2f;
typedef __attribute__((ext_vector_type(4))) float v4f;
typedef __attribute__((ext_vector_type(8))) float v8f;
typedef __attribute__((ext_vector_type(4))) unsigned int u32x4;
typedef __attribute__((ext_vector_type(4))) int i32x4;
typedef __attribute__((ext_vector_type(8))) int i32x8;

#if defined(__has_builtin)
#if __has_builtin(__builtin_amdgcn_tensor_load_to_lds) && __has_builtin(__builtin_amdgcn_s_wait_tensorcnt)
#define HAVE_TDM 1
#endif
#endif

// LDS row layout used by the attention kernel: 512 data dwords per row with a
// 2-dword pad after each 256 dwords (matches TDM pad_interval=256/pad_amount=2)
// -> row stride 516, strided lane accesses hit distinct banks.
__device__ __forceinline__ int fs_idx(int r, int c) {
  return r * 516 + c + ((c >> 8) << 1);
}

// ---------------------------------------------------------------- utilities

__global__ void k_transpose512(const float* __restrict__ in, float* __restrict__ out) {
  int i = blockIdx.x * 256 + threadIdx.x;
  int o = i >> 9, c = i & 511;
  out[c * 512 + o] = in[i];
}

__global__ void k_rownorm(const float* __restrict__ in, float* __restrict__ out) {
  __shared__ float red[128];
  int l = blockIdx.x, tid = threadIdx.x;
  float s = 0.f;
  for (int c = tid; c < C_; c += 128) { float v = in[(size_t)l * C_ + c]; s += v * v; }
  red[tid] = s; __syncthreads();
  for (int st = 64; st > 0; st >>= 1) { if (tid < st) red[tid] += red[tid + st]; __syncthreads(); }
  float inv = 1.f / fmaxf(sqrtf(red[0]), 1e-12f);
  for (int c = tid; c < C_; c += 128) out[(size_t)l * C_ + c] = in[(size_t)l * C_ + c] * inv;
}

__global__ void k_gnstats(const float* __restrict__ X, float* __restrict__ st) {
  __shared__ float r1[256], r2[256];
  int b = blockIdx.x, tid = threadIdx.x;
  const float* p = X + (size_t)b * S_ * C_;
  float s = 0.f, q = 0.f;
  for (int i = tid; i < S_ * C_; i += 256) { float v = p[i]; s += v; q += v * v; }
  r1[tid] = s; r2[tid] = q; __syncthreads();
  for (int st_ = 128; st_ > 0; st_ >>= 1) {
    if (tid < st_) { r1[tid] += r1[tid + st_]; r2[tid] += r2[tid + st_]; }
    __syncthreads();
  }
  if (tid == 0) {
    float m = r1[0] / (float)(S_ * C_);
    float var = r2[0] / (float)(S_ * C_) - m * m;
    st[b * 2] = m;
    st[b * 2 + 1] = rsqrtf(var + 1e-5f);
  }
}

// ------------------------------------------------- groupnorm + 1x1 conv GEMM
// Y[b,s,o] = (relu?)( sum_c Wt[c][o] * gn(X[b,s,c]) + bias[o] )
// grid (4 s-tiles, B), 256 threads = 8 waves; wave w owns o-chunks 4w..4w+3.
// Both WMMA operands fed from LDS; weights staged in 16-K panels (b128 loads).
__global__ __launch_bounds__(256) void k_conv_gn_wmma(
    const float* __restrict__ X, const float* __restrict__ Wt,
    const float* __restrict__ bias, const float* __restrict__ gam,
    const float* __restrict__ bet, const float* __restrict__ st,
    float* __restrict__ Y, int do_relu) {
  extern __shared__ float cm[];
  float* xt = cm;              // 16 s-rows x 512 c, stride 516
  float* wp = cm + 16 * 516;   // K-panel: 16 k-rows x 512 o, stride 516
  int b = blockIdx.y, s0 = blockIdx.x * 16, tid = threadIdx.x;
  float mean = st[b * 2], rstd = st[b * 2 + 1];

  for (int i = tid; i < 16 * 128; i += 256) {  // 2048 vec4s
    int r = i >> 7, c4 = (i & 127) << 2;
    int s = s0 + r;
    v4f v = {};
    if (s < S_) {
      v4f x = *(const v4f*)(X + ((size_t)b * S_ + s) * C_ + c4);
      v4f g = *(const v4f*)(gam + c4);
      v4f be = *(const v4f*)(bet + c4);
      v = (x - mean) * rstd * g + be;
    }
    *(v4f*)(xt + r * 516 + c4) = v;
  }

  int wave = tid >> 5, lane = tid & 31;
  int mrow = lane & 15;
  int koff = (lane < 16) ? 0 : 2;
  int ob = wave * 64 + mrow;

  v8f acc0 = {}, acc1 = {}, acc2 = {}, acc3 = {};
  for (int kp = 0; kp < 512; kp += 16) {
    __syncthreads();                      // previous panel fully consumed
    for (int i = tid; i < 16 * 128; i += 256) {
      int r = i >> 7, c4 = (i & 127) << 2;
      *(v4f*)(wp + r * 516 + c4) = *(const v4f*)(Wt + (size_t)(kp + r) * C_ + c4);
    }
    __syncthreads();
    for (int kk = 0; kk < 16; kk += 4) {
      v2f a;
      a.x = xt[mrow * 516 + kp + kk + koff];
      a.y = xt[mrow * 516 + kp + kk + koff + 1];
      const float* w0 = wp + (kk + koff) * 516;
      const float* w1 = wp + (kk + koff + 1) * 516;
      v2f b0; b0.x = w0[ob];      b0.y = w1[ob];
      v2f b1; b1.x = w0[ob + 16]; b1.y = w1[ob + 16];
      v2f b2; b2.x = w0[ob + 32]; b2.y = w1[ob + 32];
      v2f b3; b3.x = w0[ob + 48]; b3.y = w1[ob + 48];
      acc0 = __builtin_amdgcn_wmma_f32_16x16x4_f32(false, a, false, b0, (short)0, acc0, false, false);
      acc1 = __builtin_amdgcn_wmma_f32_16x16x4_f32(false, a, false, b1, (short)0, acc1, false, false);
      acc2 = __builtin_amdgcn_wmma_f32_16x16x4_f32(false, a, false, b2, (short)0, acc2, false, false);
      acc3 = __builtin_amdgcn_wmma_f32_16x16x4_f32(false, a, false, b3, (short)0, acc3, false, false);
    }
  }

  int lbase = (lane < 16) ? 0 : 8;
  v8f accs[4] = {acc0, acc1, acc2, acc3};
  for (int j = 0; j < 4; ++j) {
    int o = wave * 64 + j * 16 + mrow;
    float bi = bias[o];
    for (int e = 0; e < 8; ++e) {
      int s = s0 + e + lbase;
      if (s < S_) {
        float v = accs[j][e] + bi;
        if (do_relu) v = fmaxf(v, 0.f);
        Y[((size_t)b * S_ + s) * C_ + o] = v;
      }
    }
  }
}

// ------------------------------------------------------------- MLP (Fv path)
__device__ __forceinline__ float blockReduceSum256(float v, float* red) {
  int tid = threadIdx.x;
  __syncthreads();
  red[tid] = v; __syncthreads();
  for (int st = 128; st > 0; st >>= 1) { if (tid < st) red[tid] += red[tid + st]; __syncthreads(); }
  return red[0];
}

__device__ __forceinline__ float dot512_lds(const float* __restrict__ wrow, const float* vb) {
  float s = 0.f;
  for (int c = 0; c < 512; c += 4) {
    v4f w = *(const v4f*)(wrow + c);
    s += w.x * vb[c] + w.y * vb[c + 1] + w.z * vb[c + 2] + w.w * vb[c + 3];
  }
  return s;
}

__global__ __launch_bounds__(256) void k_mlp(
    const float* __restrict__ Fv,
    const float* __restrict__ ln1g, const float* __restrict__ ln1b,
    const float* __restrict__ w1, const float* __restrict__ b1,
    const float* __restrict__ ln2g, const float* __restrict__ ln2b,
    const float* __restrict__ w2, const float* __restrict__ b2,
    float* __restrict__ Fvp, float* __restrict__ Fvn) {
  __shared__ float vb[512];
  __shared__ float red[256];
  int b = blockIdx.x, tid = threadIdx.x;
  const float* x = Fv + (size_t)b * C_;
  float x0 = x[tid], x1 = x[tid + 256];
  float m = blockReduceSum256(x0 + x1, red) / 512.f;
  float d0 = x0 - m, d1 = x1 - m;
  float var = blockReduceSum256(d0 * d0 + d1 * d1, red) / 512.f;
  float rstd = rsqrtf(var + 1e-5f);
  __syncthreads();
  vb[tid] = d0 * rstd * ln1g[tid] + ln1b[tid];
  vb[tid + 256] = d1 * rstd * ln1g[tid + 256] + ln1b[tid + 256];
  __syncthreads();
  float h0 = b1[tid] + dot512_lds(w1 + (size_t)tid * 512, vb);
  float h1v = b1[tid + 256] + dot512_lds(w1 + (size_t)(tid + 256) * 512, vb);
  h0 = fmaxf(h0, 0.f); h1v = fmaxf(h1v, 0.f);
  m = blockReduceSum256(h0 + h1v, red) / 512.f;
  d0 = h0 - m; d1 = h1v - m;
  var = blockReduceSum256(d0 * d0 + d1 * d1, red) / 512.f;
  rstd = rsqrtf(var + 1e-5f);
  __syncthreads();
  vb[tid] = d0 * rstd * ln2g[tid] + ln2b[tid];
  vb[tid + 256] = d1 * rstd * ln2g[tid + 256] + ln2b[tid + 256];
  __syncthreads();
  float o0 = b2[tid] + dot512_lds(w2 + (size_t)tid * 512, vb);
  float o1 = b2[tid + 256] + dot512_lds(w2 + (size_t)(tid + 256) * 512, vb);
  Fvp[(size_t)b * C_ + tid] = o0;
  Fvp[(size_t)b * C_ + tid + 256] = o1;
  float n2 = blockReduceSum256(o0 * o0 + o1 * o1, red);
  float inv = 1.f / fmaxf(sqrtf(n2), 1e-12f);
  Fvn[(size_t)b * C_ + tid] = o0 * inv;
  Fvn[(size_t)b * C_ + tid + 256] = o1 * inv;
}

// ---------------------------------------------------------------- logits1
__global__ void k_logits1(const float* __restrict__ Fvn, const float* __restrict__ Ftn,
                          float* __restrict__ out) {
  __shared__ float v[512];
  int b = blockIdx.x, tid = threadIdx.x;
  v[tid] = Fvn[(size_t)b * C_ + tid];
  v[tid + 256] = Fvn[(size_t)b * C_ + tid + 256];
  __syncthreads();
  for (int l = tid; l < L_; l += 256) {
    float s = dot512_lds(Ftn + (size_t)l * C_, v);
    out[(size_t)b * L_ + l] = s;
  }
}

// ------------------------------------------ fused attention (logits2/logits3)
// grid (63 l-tiles, B), 256 threads = 8 waves: waves 0-3 = Ft attention,
// waves 4-7 = prototype attention. Fs_p[b] pulled into LDS by the TDM.
__global__ __launch_bounds__(256) void k_attn_wmma(
    const float* __restrict__ Fsp, const float* __restrict__ Ft,
    const float* __restrict__ Proto, const float* __restrict__ Ftn,
    const float* __restrict__ Proton, const float* __restrict__ Fvp,
    const float* __restrict__ gp, const float* __restrict__ gn,
    float* __restrict__ out2, float* __restrict__ out3) {
  extern __shared__ float smem[];
  float* fs = smem;                 // 52 rows (49 valid + zero pad) x stride 516
  float* ab = smem + 52 * 516;      // attention maps: [2][16][68]
  float* rb = ab + 2 * 16 * 68;     // [0..31]=norm2(att,l), [32..63]=dot(att,l)

  int b = blockIdx.y, l0 = blockIdx.x * 16, tid = threadIdx.x;
  int wave = tid >> 5, lane = tid & 31;

  // zero K-padding rows 49..51 and the reduction buffer
  for (int i = tid; i < 3 * 516; i += 256) fs[49 * 516 + i] = 0.f;
  if (tid < 64) rb[tid] = 0.f;

#if defined(HAVE_TDM)
  if (wave == 0) {
    // Tensor Data Mover: 2D tile, 4B elems, 512 x 49, LDS padding 2 dwords per
    // 256 -> row stride 516 dwords in LDS (bank-conflict-free phase-1 reads).
    unsigned lds_addr = (unsigned)__builtin_amdgcn_groupstaticsize();  // fs @ dyn-LDS base
    unsigned long long ga = (unsigned long long)(const void*)(Fsp + (size_t)b * S_ * C_);
    u32x4 g0;
    g0[0] = 1u;                                            // count=1 valid descriptor
    g0[1] = lds_addr;                                      // LDS destination
    g0[2] = (unsigned)(ga & 0xFFFFFFFFull);                // global_addr[31:0]
    g0[3] = (unsigned)((ga >> 32) & 0x01FFFFFFull) | (2u << 30);  // addr[56:32] | type=2
    i32x8 g1;
    g1[0] = (int)((2u << 16) | (1u << 20) | (7u << 22) | (1u << 25));
    //         data_size=4B | pad_en     | ivl=256dw    | pad=2dw  ; wg_mask=0
    g1[1] = (int)(512u << 16);   // tensor_dim0 = 512 (elements)
    g1[2] = (int)(49u << 16);    // tensor_dim0[31:16]=0 | tensor_dim1 = 49
    g1[3] = (int)(512u << 16);   // tensor_dim1[31:16]=0 | tile_dim0 = 512
    g1[4] = 49;                  // tile_dim1 = 49, tile_dim2 = 0
    g1[5] = 512;                 // tensor_dim0_stride lo
    g1[6] = 0;
    g1[7] = 0;
    i32x4 gz = {0, 0, 0, 0};
#if __clang_major__ >= 23
    i32x8 gz8 = {0, 0, 0, 0, 0, 0, 0, 0};
    __builtin_amdgcn_tensor_load_to_lds(g0, g1, gz, gz, gz8, 0);
#else
    __builtin_amdgcn_tensor_load_to_lds(g0, g1, gz, gz, 0);
#endif
    __builtin_amdgcn_s_wait_tensorcnt(0);
  }
#else
  for (int i = tid; i < 49 * 256; i += 256) {   // float2 cooperative fallback
    int r = i >> 8, c2 = (i & 255) << 1;
    *(float2*)(fs + fs_idx(r, c2)) =
        *(const float2*)(Fsp + ((size_t)b * S_ + r) * C_ + c2);
  }
#endif
  __syncthreads();

  int att = wave >> 2, sub = wave & 3;
  int mrow = lane & 15;
  int koff = (lane < 16) ? 0 : 2;
  int lbase = (lane < 16) ? 0 : 8;

  // phase 1: A[l,s] = sum_c Q[l,c] * Fs_p[b,s,c]   (K=512)
  {
    const float* Q = att ? Proto : Ft;
    int lq = l0 + mrow; if (lq > L_ - 1) lq = L_ - 1;
    const float* qr = Q + (size_t)lq * C_;
    int base = (sub * 16 + mrow) * 516;
    v8f acc = {};
    for (int k0 = 0; k0 < 512; k0 += 4) {
      int k = k0 + koff;
      int pi = base + k + ((k >> 8) << 1);
      v2f a;  a.x = qr[k];   a.y = qr[k + 1];
      v2f bb; bb.x = fs[pi]; bb.y = fs[pi + 1];
      acc = __builtin_amdgcn_wmma_f32_16x16x4_f32(false, a, false, bb, (short)0, acc, false, false);
    }
    float* dst = ab + att * 16 * 68;
    for (int e = 0; e < 8; ++e)
      dst[(e + lbase) * 68 + sub * 16 + mrow] = acc[e];
  }
  __syncthreads();

  // phase 2: row softmax over the 49 valid s positions; zero K-padding
  if (tid < 32) {
    float* row = ab + (tid >> 4) * 16 * 68 + (tid & 15) * 68;
    float mx = -1e30f;
    for (int s = 0; s < S_; ++s) mx = fmaxf(mx, row[s]);
    float sum = 0.f;
    for (int s = 0; s < S_; ++s) { float e = __expf(row[s] - mx); row[s] = e; sum += e; }
    float inv = 1.f / sum;
    for (int s = 0; s < S_; ++s) row[s] *= inv;
    for (int s = S_; s < 52; ++s) row[s] = 0.f;
  }
  __syncthreads();

  // phase 3: feat = A1 @ Fs_p (K=52), fused with gamma*feat + Fv_p, norm & dot
  {
    float gamma = att ? gn[0] : gp[0];
    const float* Nn = att ? Proton : Ftn;
    const float* arow = ab + att * 16 * 68 + mrow * 68;
    float pn[8] = {0.f}, pd[8] = {0.f};
    for (int j = 0; j < 8; ++j) {
      int c0 = ((wave & 3) + 4 * j) * 16;
      int cp = c0 + mrow; cp += (cp >> 8) << 1;   // padded column index
      v8f acc = {};
      for (int k0 = 0; k0 < 52; k0 += 4) {
        int k = k0 + koff;
        v2f a;  a.x = arow[k];            a.y = arow[k + 1];
        v2f bb; bb.x = fs[k * 516 + cp];  bb.y = fs[(k + 1) * 516 + cp];
        acc = __builtin_amdgcn_wmma_f32_16x16x4_f32(false, a, false, bb, (short)0, acc, false, false);
      }
      int c = c0 + mrow;
      float fvc = Fvp[(size_t)b * C_ + c];
      for (int e = 0; e < 8; ++e) {
        int lq = l0 + e + lbase; if (lq > L_ - 1) lq = L_ - 1;
        float fv = gamma * acc[e] + fvc;
        pn[e] += fv * fv;
        pd[e] += fv * Nn[(size_t)lq * C_ + c];
      }
    }
    for (int e = 0; e < 8; ++e) {
      atomicAdd(&rb[att * 16 + e + lbase], pn[e]);
      atomicAdd(&rb[32 + att * 16 + e + lbase], pd[e]);
    }
  }
  __syncthreads();

  if (tid < 32) {
    int a2 = tid >> 4, l = tid & 15, lg = l0 + l;
    if (lg < L_) {
      float n2 = rb[a2 * 16 + l], dt = rb[32 + a2 * 16 + l];
      float lv = dt / fmaxf(sqrtf(n2), 1e-12f);
      float* o = a2 ? out3 : out2;
      o[(size_t)b * L_ + lg] = lv;
    }
  }
}

// ------------------------------------------------------- prototype EMA update
__global__ void k_zero(float* __restrict__ p, int n) {
  int i = blockIdx.x * 256 + threadIdx.x;
  if (i < n) p[i] = 0.f;
}

__global__ void k_protoacc(const int* __restrict__ label, const float* __restrict__ Fvn,
                           float* __restrict__ csum, int* __restrict__ ccnt) {
  int b = blockIdx.x, tid = threadIdx.x;
  int lab = label[b];
  for (int c = tid; c < C_; c += 256)
    atomicAdd(&csum[(size_t)lab * C_ + c], Fvn[(size_t)b * C_ + c]);
  if (tid == 0) atomicAdd(&ccnt[lab], 1);
}

__global__ void k_protofinal(const float* __restrict__ proto, const float* __restrict__ csum,
                             const int* __restrict__ ccnt, float* __restrict__ out) {
  int l = blockIdx.x, tid = threadIdx.x;
  int cnt = ccnt[l];
  for (int c = tid; c < C_; c += 256) {
    float p = proto[(size_t)l * C_ + c];
    float v = p;
    if (cnt > 0)
      v = 0.99f * p + 0.01f * (csum[(size_t)l * C_ + c] / fmaxf((float)cnt, 1e-8f));
    out[(size_t)l * C_ + c] = v;
  }
}

// --------------------------------------------------------------------- launch
extern "C" void kernel_launch(void* const* d_in, const int* in_sizes, int n_in,
                              void* d_out, int out_size, void* d_ws, size_t ws_size,
                              hipStream_t stream) {
  const float* Fs    = (const float*)d_in[0];
  const float* Ft    = (const float*)d_in[1];
  const float* Fv    = (const float*)d_in[2];
  const int*   label = (const int*)d_in[3];
  const float* gn1g  = (const float*)d_in[4];
  const float* gn1bb = (const float*)d_in[5];
  const float* cw1   = (const float*)d_in[6];
  const float* cb1   = (const float*)d_in[7];
  const float* gn2g  = (const float*)d_in[8];
  const float* gn2bb = (const float*)d_in[9];
  const float* cw2   = (const float*)d_in[10];
  const float* cb2   = (const float*)d_in[11];
  const float* ln1g  = (const float*)d_in[12];
  const float* ln1b  = (const float*)d_in[13];
  const float* lw1   = (const float*)d_in[14];
  const float* lb1   = (const float*)d_in[15];
  const float* ln2g  = (const float*)d_in[16];
  const float* ln2b  = (const float*)d_in[17];
  const float* lw2   = (const float*)d_in[18];
  const float* lb2   = (const float*)d_in[19];
  const float* gp    = (const float*)d_in[20];
  const float* gnn   = (const float*)d_in[21];
  const float* proto = (const float*)d_in[22];

  float* ws  = (float*)d_ws;
  float* w1t = ws;
  float* w2t = w1t + 512 * 512;
  float* ftn = w2t + 512 * 512;
  float* ptn = ftn + (size_t)L_ * C_;
  float* h1  = ptn + (size_t)L_ * C_;
  float* fsp = h1 + (size_t)B_ * S_ * C_;
  float* fvp = fsp + (size_t)B_ * S_ * C_;
  float* fvn = fvp + (size_t)B_ * C_;
  float* st1 = fvn + (size_t)B_ * C_;
  float* st2 = st1 + 2 * B_;
  float* csum = st2 + 2 * B_;
  int*   ccnt = (int*)(csum + (size_t)L_ * C_);

  float* out  = (float*)d_out;           // (3,B,L) logits then (L,C) prototype
  float* out1 = out;
  float* out2 = out + (size_t)B_ * L_;
  float* out3 = out + 2 * (size_t)B_ * L_;
  float* outP = out + 3 * (size_t)B_ * L_;

  k_transpose512<<<1024, 256, 0, stream>>>(cw1, w1t);
  k_transpose512<<<1024, 256, 0, stream>>>(cw2, w2t);
  k_rownorm<<<L_, 128, 0, stream>>>(Ft, ftn);
  k_rownorm<<<L_, 128, 0, stream>>>(proto, ptn);

  size_t csm = (size_t)(2 * 16 * 516) * sizeof(float);
  k_gnstats<<<B_, 256, 0, stream>>>(Fs, st1);
  k_conv_gn_wmma<<<dim3(4, B_), 256, csm, stream>>>(Fs, w1t, cb1, gn1g, gn1bb, st1, h1, 1);
  k_gnstats<<<B_, 256, 0, stream>>>(h1, st2);
  k_conv_gn_wmma<<<dim3(4, B_), 256, csm, stream>>>(h1, w2t, cb2, gn2g, gn2bb, st2, fsp, 0);

  k_mlp<<<B_, 256, 0, stream>>>(Fv, ln1g, ln1b, lw1, lb1, ln2g, ln2b, lw2, lb2, fvp, fvn);
  k_logits1<<<B_, 256, 0, stream>>>(fvn, ftn, out1);

  size_t asm_ = (size_t)(52 * 516 + 2 * 16 * 68 + 64) * sizeof(float);
  k_attn_wmma<<<dim3(63, B_), 256, asm_, stream>>>(fsp, Ft, proto, ftn, ptn, fvp, gp, gnn,
                                                   out2, out3);

  int nz = L_ * C_ + L_;  // csum + ccnt (zero bits valid for both float and int)
  k_zero<<<(nz + 255) / 256, 256, 0, stream>>>(csum, nz);
  k_protoacc<<<B_, 256, 0, stream>>>(label, fvn, csum, ccnt);
  k_protofinal<<<L_, 256, 0, stream>>>(proto, csum, ccnt, outP);
}